// Circuit_32693291057889
// MI455X (gfx1250) — compile-verified
//
#include <hip/hip_runtime.h>
#include <stdint.h>

// Problem constants (fixed by the reference's setup_inputs()).
#define NV     10000
#define NC     42000
#define BATCH  1024
#define KLIT   3

// Broadcast tiling: 4096-column tiles (16 KB LDS), 16 rows per block.
#define TC     4096
#define ROWS   16

// ---------------------------------------------------------------------------
// Kernel 1: per-clause soft-OR.  r[c] = 1 - prod_k (1 - y_k),
//   y_k = sign>0 ? sigmoid(emb[idx]) : 1 - sigmoid(emb[idx]).
// input_idx is always 0 (emb has exactly one row), so the batch dimension
// collapses and we compute the [NC] vector once.
// ---------------------------------------------------------------------------
__global__ void __launch_bounds__(256)
clause_eval_kernel(const float* __restrict__ emb,
                   const int*   __restrict__ cidx,
                   const float* __restrict__ csign,
                   float*       __restrict__ r) {
    int c = blockIdx.x * blockDim.x + threadIdx.x;
    if (c >= NC) return;
    float prod = 1.0f;
#pragma unroll
    for (int k = 0; k < KLIT; ++k) {
        int   v = cidx[c * KLIT + k];
        float s = csign[c * KLIT + k];
        float w = emb[v];
        float p = 1.0f / (1.0f + expf(-w));     // sigmoid
        float y = (s > 0.0f) ? p : (1.0f - p);  // signed literal
        prod *= (1.0f - y);
    }
    r[c] = 1.0f - prod;
}

// ---------------------------------------------------------------------------
// Kernel 2: broadcast r[NC] to out[BATCH][NC].
// Stage the column tile into LDS with gfx1250 async global->LDS DMA, then
// stream it to ROWS rows with aligned 128-bit stores.  Pure store-BW bound:
// 172 MB of stores -> ~7.4 us floor at 23.3 TB/s.
// ---------------------------------------------------------------------------
__global__ void __launch_bounds__(256)
broadcast_kernel(const float* __restrict__ r, float* __restrict__ out) {
    __shared__ __align__(16) float tile[TC];

    const int c0     = blockIdx.x * TC;
    const int tileN  = min(TC, NC - c0);   // 4096, or 1040 for the tail tile
    const int groups = tileN >> 2;         // float4 groups in this tile
    const int tid    = threadIdx.x;

    // Low 32 bits of a flat shared-aperture pointer == LDS byte offset.
    unsigned ldsbase = (unsigned)(uintptr_t)(&tile[0]);

    // Async global -> LDS copy of this block's slice of r (ASYNCcnt path).
#pragma unroll
    for (int j = 0; j < TC / 4 / 256; ++j) {
        int g = tid + j * 256;
        if (g < groups) {
            uint64_t gaddr = (uint64_t)(uintptr_t)(r + c0 + 4 * g);
            unsigned laddr = ldsbase + 16u * (unsigned)g;
            asm volatile("global_load_async_to_lds_b128 %0, %1, off"
                         :: "v"(laddr), "v"(gaddr)
                         : "memory");
        }
    }
    asm volatile("s_wait_asynccnt 0" ::: "memory");
    __syncthreads();

    // Hoist this thread's float4 groups into registers (constant indices,
    // fully unrolled -> no scratch spills).
    float4 vals[TC / 4 / 256];
#pragma unroll
    for (int j = 0; j < TC / 4 / 256; ++j) {
        int g = tid + j * 256;
        if (g < groups)
            vals[j] = *reinterpret_cast<const float4*>(&tile[4 * g]);
    }

    // Fan out to ROWS rows.  Row stride NC*4 = 168000 B and all column
    // offsets are multiples of 16 B -> every store is an aligned b128.
    const int row0 = blockIdx.y * ROWS;
    for (int row = row0; row < row0 + ROWS; ++row) {
        float* base = out + (size_t)row * NC + c0;
#pragma unroll
        for (int j = 0; j < TC / 4 / 256; ++j) {
            int g = tid + j * 256;
            if (g < groups)
                *reinterpret_cast<float4*>(base + 4 * g) = vals[j];
        }
    }
}

// ---------------------------------------------------------------------------
// Launch: inputs in setup_inputs() order:
//   d_in[0] input_idx  (int64[B])   -- unused: emb has 1 row, all indices == 0
//   d_in[1] emb_weight (f32[1,NV])
//   d_in[2] clause_idx (i32[NC,K])
//   d_in[3] clause_sign(f32[NC,K])
// d_out: f32[B, NC].  d_ws: r[NC] scratch (168 KB).
// ---------------------------------------------------------------------------
extern "C" void kernel_launch(void* const* d_in, const int* in_sizes, int n_in,
                              void* d_out, int out_size, void* d_ws, size_t ws_size,
                              hipStream_t stream) {
    const float* emb   = (const float*)d_in[1];
    const int*   cidx  = (const int*)  d_in[2];
    const float* csign = (const float*)d_in[3];
    float*       r     = (float*)d_ws;
    float*       out   = (float*)d_out;

    clause_eval_kernel<<<(NC + 255) / 256, 256, 0, stream>>>(emb, cidx, csign, r);

    dim3 grid((NC + TC - 1) / TC, BATCH / ROWS);   // 11 x 64 = 704 blocks
    broadcast_kernel<<<grid, 256, 0, stream>>>(r, out);
}